// ZCA2_37185826849259
// MI455X (gfx1250) — compile-verified
//
#include <hip/hip_runtime.h>

typedef float v2f __attribute__((ext_vector_type(2)));
typedef float v8f __attribute__((ext_vector_type(8)));

#define ZEPS 1e-5f
#define NS_ITERS 5

// Problem geometry (fixed by the reference)
#define BSZ 128
#define CSZ 256
#define HWSZ 1024
#define NCH 64
#define NGRP 4
#define MTOT (BSZ*HWSZ)      // 131072

// Workspace layout (in floats)
#define WS_SUM 0             // 4*64
#define WS_S   256           // 4*64*64 (upper-tri tiles filled)
#define WS_WM  16640         // 4*64*64 whitening matrices
#define WS_OFF 33024         // 4*64   wm @ mu

// ---------------------------------------------------------------------------
__global__ void zca_init(float* ws) {
    int t = blockIdx.x * blockDim.x + threadIdx.x;
    if (t < 16640) ws[t] = 0.0f;
}

// ---------------------------------------------------------------------------
// Kernel 1: per-channel sums + S = x x^T (per group), f32 WMMA accumulation.
// Grid: (128 batches, 4 groups), 256 threads (8 waves).
__global__ __launch_bounds__(256) void zca_cov(const float* __restrict__ X,
                                               float* __restrict__ ws) {
    __shared__ float lds[64 * 132];          // stage tile 64ch x 128m, pitch 132
    const int t    = threadIdx.x;
    const int wave = t >> 5;
    const int lane = t & 31;
    const int c0   = lane & 15;
    const int half = lane >> 4;
    const int b    = blockIdx.x;
    const int g    = blockIdx.y;

    const float* xg = X + (size_t)b * (CSZ * HWSZ) + g * (NCH * HWSZ);

    const int TI[10] = {0,0,0,0,1,1,1,2,2,3};
    const int TJ[10] = {0,1,2,3,1,2,3,2,3,3};

    v8f acc[10];
#pragma unroll
    for (int p = 0; p < 10; ++p) acc[p] = (v8f)(0.0f);

    float csum = 0.0f;

    for (int chunk = 0; chunk < 8; ++chunk) {
        const int hw0 = chunk * 128;
        __syncthreads();
        // stage: each wave fills rows (wave + 8*i), 32 lanes x float4 = 512B/row
        {
            const int colb = (t & 31) * 4;
            const int row0 = t >> 5;
#pragma unroll
            for (int i = 0; i < 8; ++i) {
                const int row = row0 + i * 8;
                const float4 v = *(const float4*)(xg + row * HWSZ + hw0 + colb);
                *(float4*)&lds[row * 132 + colb] = v;
            }
        }
        __syncthreads();
        // per-channel partial sums (waves 0-1)
        if (t < 64) {
            float s = 0.0f;
#pragma unroll 8
            for (int j = 0; j < 128; ++j) s += lds[t * 132 + j];
            csum += s;
        }
        // WMMA: wave handles K4-steps {wave + 8*s}; A and B layouts coincide
        // for x*x^T, so 4 ds_load_b64 feed 10 upper-tri tile WMMAs.
#pragma unroll
        for (int s = 0; s < 4; ++s) {
            const int k0 = (wave + 8 * s) * 4;
            v2f a[4];
#pragma unroll
            for (int ct = 0; ct < 4; ++ct)
                a[ct] = *(const v2f*)&lds[(ct * 16 + c0) * 132 + k0 + 2 * half];
#pragma unroll
            for (int p = 0; p < 10; ++p)
                acc[p] = __builtin_amdgcn_wmma_f32_16x16x4_f32(
                    false, a[TI[p]], false, a[TJ[p]], (short)0, acc[p], false, false);
        }
    }
    __syncthreads();
    // cross-wave reduce: reuse first 4096 floats of lds as a 64x64 buffer
    for (int e = t; e < 4096; e += 256) lds[e] = 0.0f;
    __syncthreads();
#pragma unroll
    for (int p = 0; p < 10; ++p) {
#pragma unroll
        for (int v = 0; v < 8; ++v) {
            const int row = TI[p] * 16 + v + 8 * half;
            const int col = TJ[p] * 16 + c0;
            atomicAdd(&lds[row * 64 + col], acc[p][v]);
        }
    }
    __syncthreads();
    float* Sg = ws + WS_S + g * 4096;
#pragma unroll
    for (int p = 0; p < 10; ++p) {
        const int row = TI[p] * 16 + (t >> 4);
        const int col = TJ[p] * 16 + (t & 15);
        atomicAdd(&Sg[row * 64 + col], lds[row * 64 + col]);
    }
    if (t < 64) atomicAdd(&ws[WS_SUM + g * 64 + t], csum);
}

// ---------------------------------------------------------------------------
// Kernel 2 helpers: 16x16 tile matmul out of LDS (pitch 66) via f32 WMMA.
__device__ __forceinline__ v8f mm_tile(const float* A, const float* Bm,
                                       int ti, int tj, int c0, int half) {
    v8f acc = (v8f)(0.0f);
#pragma unroll
    for (int kk = 0; kk < 16; ++kk) {
        const int k0 = kk * 4 + 2 * half;
        v2f a = *(const v2f*)&A[(ti * 16 + c0) * 66 + k0];
        v2f b;
        b.x = Bm[(k0    ) * 66 + tj * 16 + c0];
        b.y = Bm[(k0 + 1) * 66 + tj * 16 + c0];
        acc = __builtin_amdgcn_wmma_f32_16x16x4_f32(false, a, false, b,
                                                    (short)0, acc, false, false);
    }
    return acc;
}

__device__ __forceinline__ void st_tile(float* D, v8f r, int ti, int tj,
                                        int c0, int half) {
#pragma unroll
    for (int v = 0; v < 8; ++v)
        D[(ti * 16 + v + 8 * half) * 66 + tj * 16 + c0] = r[v];
}

// Kernel 2: finalize Sigma, Newton-Schulz (T=5), emit wm and off = wm@mu.
// Grid: 4 blocks (one per group), 512 threads = 16 waves (4x4 tile grid).
__global__ __launch_bounds__(512) void zca_ns(float* __restrict__ ws) {
    __shared__ float Sn[64 * 66];
    __shared__ float P [64 * 66];
    __shared__ float T1[64 * 66];
    __shared__ float mu[64];
    __shared__ float red;

    const int g    = blockIdx.x;
    const int t    = threadIdx.x;
    const int wave = t >> 5;
    const int lane = t & 31;
    const int c0   = lane & 15;
    const int half = lane >> 4;
    const int ti   = wave >> 2;
    const int tj   = wave & 3;

    const float invM = 1.0f / (float)MTOT;
    const float* Ssrc = ws + WS_S + g * 4096;

    if (t < 64) mu[t] = ws[WS_SUM + g * 64 + t] * invM;
    if (t == 0) red = 0.0f;
    __syncthreads();

    for (int e = t; e < 4096; e += 512) {
        const int i = e >> 6, j = e & 63;
        const int ii = min(i, j), jj = max(i, j);   // S holds upper-tri tiles
        float v = Ssrc[ii * 64 + jj] * invM - mu[i] * mu[j]
                + ((i == j) ? ZEPS : 0.0f);
        Sn[i * 66 + j] = v;
        P [i * 66 + j] = (i == j) ? 1.0f : 0.0f;
    }
    __syncthreads();
    if (t < 64) atomicAdd(&red, Sn[t * 66 + t]);
    __syncthreads();
    const float rTr = 1.0f / red;
    for (int e = t; e < 4096; e += 512) {
        const int i = e >> 6, j = e & 63;
        Sn[i * 66 + j] *= rTr;
    }
    __syncthreads();

    for (int it = 0; it < NS_ITERS; ++it) {
        v8f r = mm_tile(P, P, ti, tj, c0, half);      // P2
        st_tile(T1, r, ti, tj, c0, half);
        __syncthreads();
        r = mm_tile(T1, P, ti, tj, c0, half);         // P3
        __syncthreads();                              // reads of T1 done
        st_tile(T1, r, ti, tj, c0, half);
        __syncthreads();
        r = mm_tile(T1, Sn, ti, tj, c0, half);        // P3 @ Sn
        v8f pn;
#pragma unroll
        for (int v = 0; v < 8; ++v) {
            const float pv = P[(ti * 16 + v + 8 * half) * 66 + tj * 16 + c0];
            pn[v] = 1.5f * pv - 0.5f * r[v];
        }
        __syncthreads();                              // reads of P done
        st_tile(P, pn, ti, tj, c0, half);
        __syncthreads();
    }

    const float srt = sqrtf(rTr);
    float* WMg = ws + WS_WM + g * 4096;
    for (int e = t; e < 4096; e += 512) {
        const int i = e >> 6, j = e & 63;
        WMg[i * 64 + j] = P[i * 66 + j] * srt;
    }
    if (t < 64) {
        float o = 0.0f;
#pragma unroll 8
        for (int d = 0; d < 64; ++d) o += P[t * 66 + d] * mu[d];
        ws[WS_OFF + g * 64 + t] = o * srt;
    }
}

// ---------------------------------------------------------------------------
// Kernel 3: out = (wm @ x - off) * weight + bias, streamed by f32 WMMA.
// Grid: (256, 4 groups), 256 threads = 8 waves. Wave = (row-half, m-tile lane),
// wm half held in VGPRs for the whole kernel; grid-stride over 16-wide m tiles.
__global__ __launch_bounds__(256) void zca_apply(const float* __restrict__ X,
                                                 const float* __restrict__ weight,
                                                 const float* __restrict__ bias,
                                                 const float* __restrict__ ws,
                                                 float* __restrict__ out) {
    const int t    = threadIdx.x;
    const int wave = t >> 5;
    const int lane = t & 31;
    const int c0   = lane & 15;
    const int half = lane >> 4;
    const int g    = blockIdx.y;
    const int rh   = wave >> 2;               // row half (channels 0-31 / 32-63)
    const float* WMg = ws + WS_WM + g * 4096;

    v2f a[2][16];                             // A operands: 2 row-tiles x 16 K-tiles
#pragma unroll
    for (int i2 = 0; i2 < 2; ++i2) {
        const int ti = rh * 2 + i2;
#pragma unroll
        for (int kk = 0; kk < 16; ++kk)
            a[i2][kk] = *(const v2f*)&WMg[(ti * 16 + c0) * 64 + kk * 4 + 2 * half];
    }
    float Wc[2][8], Bc[2][8];
#pragma unroll
    for (int i2 = 0; i2 < 2; ++i2) {
        const int ti = rh * 2 + i2;
#pragma unroll
        for (int v = 0; v < 8; ++v) {
            const int c = g * 64 + ti * 16 + v + 8 * half;
            const float w = weight[c];
            Wc[i2][v] = w;
            Bc[i2][v] = bias[c] - ws[WS_OFF + c] * w;
        }
    }

    const int tile0 = blockIdx.x * 4 + (wave & 3);
    for (int tile = tile0; tile < MTOT / 16; tile += 1024) {
        const int m  = tile * 16;
        const int b  = m >> 10;
        const int hw = m & 1023;
        const float* xb = X + (size_t)b * (CSZ * HWSZ) + g * (NCH * HWSZ) + hw + c0;
        v8f acc0 = (v8f)(0.0f), acc1 = (v8f)(0.0f);
#pragma unroll
        for (int kk = 0; kk < 16; ++kk) {
            const int d = kk * 4 + 2 * half;
            v2f bv;
            bv.x = xb[(d    ) * HWSZ];
            bv.y = xb[(d + 1) * HWSZ];
            acc0 = __builtin_amdgcn_wmma_f32_16x16x4_f32(false, a[0][kk], false, bv,
                                                         (short)0, acc0, false, false);
            acc1 = __builtin_amdgcn_wmma_f32_16x16x4_f32(false, a[1][kk], false, bv,
                                                         (short)0, acc1, false, false);
        }
        float* ob = out + (size_t)b * (CSZ * HWSZ) + g * (NCH * HWSZ) + hw + c0;
#pragma unroll
        for (int v = 0; v < 8; ++v) {
            const int r0 = (rh * 2 + 0) * 16 + v + 8 * half;
            const int r1 = (rh * 2 + 1) * 16 + v + 8 * half;
            ob[r0 * HWSZ] = acc0[v] * Wc[0][v] + Bc[0][v];
            ob[r1 * HWSZ] = acc1[v] * Wc[1][v] + Bc[1][v];
        }
    }
}

// ---------------------------------------------------------------------------
extern "C" void kernel_launch(void* const* d_in, const int* in_sizes, int n_in,
                              void* d_out, int out_size, void* d_ws, size_t ws_size,
                              hipStream_t stream) {
    (void)in_sizes; (void)n_in; (void)out_size; (void)ws_size;
    const float* X      = (const float*)d_in[0];
    const float* weight = (const float*)d_in[1];
    const float* bias   = (const float*)d_in[2];
    float* out = (float*)d_out;
    float* ws  = (float*)d_ws;

    zca_init <<<dim3(66),       dim3(256), 0, stream>>>(ws);
    zca_cov  <<<dim3(BSZ, NGRP), dim3(256), 0, stream>>>(X, ws);
    zca_ns   <<<dim3(NGRP),     dim3(512), 0, stream>>>(ws);
    zca_apply<<<dim3(256, NGRP), dim3(256), 0, stream>>>(X, weight, bias, ws, out);
}